// RotInvariantAttention_18502719111806
// MI455X (gfx1250) — compile-verified
//
#include <hip/hip_runtime.h>
#include <math.h>

// Problem constants (from reference): B=16, S=2048, K=2048, D=512 everywhere.
#define BB 16
#define SS 2048
#define DD 512
#define KKEYS 2048.0f

typedef __attribute__((ext_vector_type(2))) float v2f;
typedef __attribute__((ext_vector_type(8))) float v8f;

// ---------------------------------------------------------------------------
// Kernel 1: wqs[i] = sum_h Wq[i,h] * Ws[h]   (512 rows, one wave per row)
// ---------------------------------------------------------------------------
__global__ void k_wqs(const float* __restrict__ Wq, const float* __restrict__ Ws,
                      float* __restrict__ wqs) {
    const int row  = blockIdx.x;      // 512
    const int lane = threadIdx.x;     // 32
    float acc = 0.0f;
    for (int h = lane; h < DD; h += 32)
        acc = fmaf(Wq[row * DD + h], Ws[h], acc);
    for (int off = 16; off > 0; off >>= 1)
        acc += __shfl_xor(acc, off, 32);
    if (lane == 0) wqs[row] = acc;
}

// ---------------------------------------------------------------------------
// Kernel 2: qs[r] = x[r,:] . wqs  for r in [0, B*S)   (wave per row, float4)
// ---------------------------------------------------------------------------
__global__ void k_qs(const float* __restrict__ x, const float* __restrict__ wqs,
                     float* __restrict__ qs) {
    const int lane = threadIdx.x & 31;
    const int wave = threadIdx.x >> 5;              // 8 waves / block
    const int row  = blockIdx.x * 8 + wave;         // B*S rows total
    const float4* xr = (const float4*)(x + (size_t)row * DD);
    const float4* w4 = (const float4*)wqs;
    float acc = 0.0f;
#pragma unroll
    for (int j = 0; j < 4; ++j) {                   // 128 float4 per row
        float4 a = xr[lane + j * 32];
        float4 b = w4[lane + j * 32];
        acc = fmaf(a.x, b.x, acc);
        acc = fmaf(a.y, b.y, acc);
        acc = fmaf(a.z, b.z, acc);
        acc = fmaf(a.w, b.w, acc);
    }
    for (int off = 16; off > 0; off >>= 1)
        acc += __shfl_xor(acc, off, 32);
    if (lane == 0) qs[row] = acc;
}

// ---------------------------------------------------------------------------
// Kernel 3: per-batch softmax over S=2048 of qs/sqrt(2048) -> p
// ---------------------------------------------------------------------------
__global__ void k_softmax(const float* __restrict__ qs, float* __restrict__ p) {
    __shared__ float red[256];
    const int b = blockIdx.x;         // 16
    const int t = threadIdx.x;        // 256
    const float inv = 0.022097086912079612f;   // 1/sqrt(2048)
    float vals[8];
    float m = -INFINITY;
#pragma unroll
    for (int j = 0; j < 8; ++j) {
        vals[j] = qs[b * SS + t + j * 256] * inv;
        m = fmaxf(m, vals[j]);
    }
    red[t] = m; __syncthreads();
    for (int s = 128; s > 0; s >>= 1) {
        if (t < s) red[t] = fmaxf(red[t], red[t + s]);
        __syncthreads();
    }
    m = red[0]; __syncthreads();
    float sum = 0.0f;
#pragma unroll
    for (int j = 0; j < 8; ++j) {
        vals[j] = expf(vals[j] - m);
        sum += vals[j];
    }
    red[t] = sum; __syncthreads();
    for (int s = 128; s > 0; s >>= 1) {
        if (t < s) red[t] += red[t + s];
        __syncthreads();
    }
    const float rsum = 1.0f / red[0];
#pragma unroll
    for (int j = 0; j < 8; ++j)
        p[b * SS + t + j * 256] = vals[j] * rsum;
}

// ---------------------------------------------------------------------------
// Kernel 4: partial weighted row-sums of x: part[(b,chunk),i] =
//           sum_{q in chunk} p[b,q] * x[b,q,i]     (32 q-chunks of 64)
// ---------------------------------------------------------------------------
__global__ void k_xw_part(const float* __restrict__ x, const float* __restrict__ p,
                          float* __restrict__ part) {
    const int chunk = blockIdx.x;     // 32
    const int b     = blockIdx.y;     // 16
    const int i     = threadIdx.x;    // 512
    const int q0    = chunk * 64;
    const float* xb = x + ((size_t)b * SS + q0) * DD;
    const float* pb = p + b * SS + q0;
    float acc = 0.0f;
    for (int q = 0; q < 64; ++q)
        acc = fmaf(pb[q], xb[(size_t)q * DD + i], acc);
    part[((b * 32 + chunk) * DD) + i] = acc;
}

// ---------------------------------------------------------------------------
// Kernel 5: xw[b,i] = sum_chunk part[(b,chunk),i]
// ---------------------------------------------------------------------------
__global__ void k_xw_reduce(const float* __restrict__ part, float* __restrict__ xw) {
    const int idx = blockIdx.x * blockDim.x + threadIdx.x;  // 8192
    const int b = idx >> 9, i = idx & 511;
    float acc = 0.0f;
#pragma unroll
    for (int c = 0; c < 32; ++c)
        acc += part[((b * 32 + c) * DD) + i];
    xw[idx] = acc;
}

// ---------------------------------------------------------------------------
// Kernel 6/7: D(16x512) = scale * A(16x512) @ W(512x512)  via
//             V_WMMA_F32_16X16X4_F32 (full f32 matrix path, exact accumulate).
// One wave per block (EXEC all ones), one 16-column tile per block.
// A-layout (16x4 f32): lanes 0-15 K={0,1}, lanes 16-31 K={2,3}.
// B-layout (4x16 f32): N = lane&15 across lanes, K split by lane half.
// C/D-layout (16x16 f32): n = lane&15, VGPR r -> m = r + 8*(lane>>4).
// ---------------------------------------------------------------------------
__global__ void k_gemm16_wmma(const float* __restrict__ A, const float* __restrict__ W,
                              float* __restrict__ D, float scale) {
    const int col0 = blockIdx.x * 16;   // 32 blocks -> 512 cols
    const int lane = threadIdx.x;       // 32 threads (one wave)
    const int half = lane >> 4;
    const int mn   = lane & 15;
    v8f c = {};
    const float* Arow = A + mn * DD;    // A row m = mn for this lane
    for (int kk = 0; kk < DD; kk += 4) {
        v2f a, bv;
        a.x  = Arow[kk + 2 * half];
        a.y  = Arow[kk + 2 * half + 1];
        bv.x = W[(kk + 2 * half)     * DD + col0 + mn];
        bv.y = W[(kk + 2 * half + 1) * DD + col0 + mn];
        c = __builtin_amdgcn_wmma_f32_16x16x4_f32(
                /*neg_a=*/false, a, /*neg_b=*/false, bv,
                /*c_mod=*/(short)0, c, /*reuse_a=*/false, /*reuse_b=*/false);
    }
#pragma unroll
    for (int r = 0; r < 8; ++r) {
        const int m = r + 8 * half;
        D[m * DD + col0 + mn] = scale * c[r];
    }
}

// ---------------------------------------------------------------------------
// Launch: inputs are x, node_feats, Wq, Wk, Wv, Ws, Wo.
// node_feats/Wk provably cancel out of the softmax -> never read (saves 65 MB).
// ---------------------------------------------------------------------------
extern "C" void kernel_launch(void* const* d_in, const int* in_sizes, int n_in,
                              void* d_out, int out_size, void* d_ws, size_t ws_size,
                              hipStream_t stream) {
    const float* x  = (const float*)d_in[0];
    const float* Wq = (const float*)d_in[2];
    const float* Wv = (const float*)d_in[4];
    const float* Ws = (const float*)d_in[5];
    const float* Wo = (const float*)d_in[6];
    float* out = (float*)d_out;

    float* ws   = (float*)d_ws;
    float* wqs  = ws;                       // 512
    float* qs   = wqs + DD;                 // 32768
    float* p    = qs + BB * SS;             // 32768
    float* part = p + BB * SS;              // 16*32*512 = 262144
    float* xw   = part + BB * 32 * DD;      // 8192
    float* t1   = xw + BB * DD;             // 8192   (total ~1.35 MB)

    k_wqs      <<<DD, 32, 0, stream>>>(Wq, Ws, wqs);
    k_qs       <<<(BB * SS) / 8, 256, 0, stream>>>(x, wqs, qs);
    k_softmax  <<<BB, 256, 0, stream>>>(qs, p);
    k_xw_part  <<<dim3(32, BB), DD, 0, stream>>>(x, p, part);
    k_xw_reduce<<<32, 256, 0, stream>>>(part, xw);
    k_gemm16_wmma<<<DD / 16, 32, 0, stream>>>(xw, Wv, t1, 1.0f);
    k_gemm16_wmma<<<DD / 16, 32, 0, stream>>>(t1, Wo, out, KKEYS);
}